// RegRagContrastiveWeights_34806414966874
// MI455X (gfx1250) — compile-verified
//
#include <hip/hip_runtime.h>

// ---------------- problem constants (from reference) ----------------
#define B_IMG 2
#define D_DIM 32
#define HW 262144            // 512*512
#define C_SEG 512
#define E_EDGE 2048
#define NMAPS 4              // {emb x sp_seg, remb x rot_sp} x B
#define DELTA_VARF 0.1f
#define DELTA_DISTF 0.3f
#define EPSF 1e-10f

// ---------------- tiling ----------------
#define TPB 256
#define TILE_PX 256
#define TILE_FLOATS (TILE_PX * D_DIM)          // 8192 floats = 32KB
#define BLOCKS_PER_MAP 64
#define TILES_PER_BLOCK (HW / (TILE_PX * BLOCKS_PER_MAP))  // 16

// ---------------- workspace layout (floats) ----------------
#define OFF_SUM 0                                  // NMAPS*C*D = 65536
#define OFF_CNT (NMAPS * C_SEG * D_DIM)            // 65536 (+2048)
#define OFF_SCAL (OFF_CNT + NMAPS * C_SEG)         // 67584 (+16)
#define OFF_MEAN (OFF_SCAL + 16)                   // 67600 (+65536)
#define ZERO_FLOATS OFF_MEAN

typedef __attribute__((ext_vector_type(2))) float v2f;
typedef __attribute__((ext_vector_type(8))) float v8f;
typedef __attribute__((ext_vector_type(4))) int v4i;

// ---------------- CDNA5 async global->LDS path (guarded) ----------------
#if defined(__has_builtin)
#if __has_builtin(__builtin_amdgcn_global_load_async_to_lds_b128)
#define HAVE_ASYNC 1
#endif
#endif
#ifndef HAVE_ASYNC
#define HAVE_ASYNC 0
#endif

#if HAVE_ASYNC
#if __has_builtin(__builtin_amdgcn_s_wait_asynccnt)
#define WAIT_ASYNC(n) __builtin_amdgcn_s_wait_asynccnt(n)
#else
#define WAIT_ASYNC(n) asm volatile("s_wait_asynccnt %0" ::"i"(n) : "memory")
#endif
#else
#define WAIT_ASYNC(n) do { } while (0)
#endif

__device__ __forceinline__ void g2l_b128(const float* g, float* l) {
#if HAVE_ASYNC
  // builtin signature (from clang diagnostic): (v4i addrspace(1)*, v4i addrspace(3)*, imm offset, imm cpol)
  __attribute__((address_space(1))) float* gas =
      (__attribute__((address_space(1))) float*)(float*)g;
  __attribute__((address_space(3))) float* las =
      (__attribute__((address_space(3))) float*)l;
  __attribute__((address_space(1))) v4i* gv =
      (__attribute__((address_space(1))) v4i*)gas;
  __attribute__((address_space(3))) v4i* lv =
      (__attribute__((address_space(3))) v4i*)las;
  __builtin_amdgcn_global_load_async_to_lds_b128(gv, lv, 0, 0);
#else
  *(float4*)l = *(const float4*)g;
#endif
}

// stage one 256px x 32dim tile (32KB). 8 b128 async ops per thread (=8 per wave).
__device__ __forceinline__ void issue_tile(const float* ebase, int p0,
                                           float* lbuf, int t) {
#pragma unroll
  for (int q = 0; q < 8; ++q) {
    int f = q * (TPB * 4) + t * 4;      // float index in tile [d*256 + pcol]
    int d = f >> 8;
    int pcol = f & (TILE_PX - 1);
    g2l_b128(ebase + (size_t)d * HW + p0 + pcol, lbuf + f);
  }
}

// ---------------- kernel 0: zero accumulators ----------------
__global__ void k_zero(float* __restrict__ ws) {
  int i = blockIdx.x * blockDim.x + threadIdx.x;
  if (i < ZERO_FLOATS) ws[i] = 0.0f;
}

// ---------------- kernel 1: segmented sums + counts ----------------
__global__ __launch_bounds__(TPB) void k_segsum(const float* __restrict__ emb,
                                                const int* __restrict__ sp,
                                                const int* __restrict__ rsp,
                                                float* __restrict__ ws) {
  __shared__ __align__(16) float lsum[C_SEG * D_DIM];   // 64KB
  __shared__ float lcnt[C_SEG];                         // 2KB
  __shared__ __align__(16) float ltile[2][TILE_FLOATS]; // 64KB double buffer

  const int n = blockIdx.x / BLOCKS_PER_MAP;
  const int blk = blockIdx.x % BLOCKS_PER_MAP;
  const int t = threadIdx.x;
  const float* ebase = emb + (size_t)n * D_DIM * HW;
  const int* seg = (n < B_IMG) ? (sp + (size_t)n * HW)
                               : (rsp + (size_t)(n - B_IMG) * HW);
  const int pix0 = blk * (TILES_PER_BLOCK * TILE_PX);

  for (int i = t; i < C_SEG * D_DIM; i += TPB) lsum[i] = 0.0f;
  for (int i = t; i < C_SEG; i += TPB) lcnt[i] = 0.0f;

  issue_tile(ebase, pix0, ltile[0], t);

  for (int tile = 0; tile < TILES_PER_BLOCK; ++tile) {
    if (tile + 1 < TILES_PER_BLOCK) {
      issue_tile(ebase, pix0 + (tile + 1) * TILE_PX, ltile[(tile + 1) & 1], t);
      WAIT_ASYNC(8);     // previous tile's 8 per-wave async ops complete
    } else {
      WAIT_ASYNC(0);
    }
    __syncthreads();

    const float* buf = ltile[tile & 1];
    const int p = pix0 + tile * TILE_PX + t;
    const int s = seg[p];
    atomicAdd(&lcnt[s], 1.0f);
#pragma unroll
    for (int d = 0; d < D_DIM; ++d)
      atomicAdd(&lsum[s * D_DIM + d], buf[d * TILE_PX + t]);
    __syncthreads();   // protect buffer reuse two iterations ahead
  }

  float* gsum = ws + OFF_SUM + (size_t)n * C_SEG * D_DIM;
  for (int i = t; i < C_SEG * D_DIM; i += TPB) atomicAdd(&gsum[i], lsum[i]);
  float* gcnt = ws + OFF_CNT + n * C_SEG;
  for (int i = t; i < C_SEG; i += TPB) atomicAdd(&gcnt[i], lcnt[i]);
}

// ---------------- kernel 2: normalize means (per space, +EPS after div) ----
__global__ void k_means(float* __restrict__ ws) {
  int t = blockIdx.x * blockDim.x + threadIdx.x;  // t = n*C + c
  if (t >= NMAPS * C_SEG) return;
  const float* s = ws + OFF_SUM + (size_t)t * D_DIM;
  const float cnt = ws[OFF_CNT + t];
  const float inv = 1.0f / cnt;
  float m[D_DIM];
  float n0 = 0.0f, n1 = 0.0f;
#pragma unroll
  for (int d = 0; d < 16; ++d) { m[d] = s[d] * inv; n0 += m[d] * m[d]; }
#pragma unroll
  for (int d = 16; d < 32; ++d) { m[d] = s[d] * inv; n1 += m[d] * m[d]; }
  n0 = 1.0f / sqrtf(n0);
  n1 = 1.0f / sqrtf(n1);
  float* o = ws + OFF_MEAN + (size_t)t * D_DIM;
#pragma unroll
  for (int d = 0; d < 16; ++d) o[d] = m[d] * n0 + EPSF;
#pragma unroll
  for (int d = 16; d < 32; ++d) o[d] = m[d] * n1 + EPSF;
}

// ---------------- kernel 3: intra loss (2nd streaming pass) ----------------
#define MSTRIDE 36   // padded LDS row stride (floats): breaks bank conflicts
__global__ __launch_bounds__(TPB) void k_intra(const float* __restrict__ emb,
                                               const int* __restrict__ sp,
                                               const int* __restrict__ rsp,
                                               float* __restrict__ ws) {
  __shared__ __align__(16) float lmean[C_SEG * MSTRIDE]; // 72KB
  __shared__ float lcnt[C_SEG];
  __shared__ __align__(16) float ltile[2][TILE_FLOATS];  // 64KB
  __shared__ float lred;

  const int n = blockIdx.x / BLOCKS_PER_MAP;
  const int blk = blockIdx.x % BLOCKS_PER_MAP;
  const int t = threadIdx.x;
  const float* ebase = emb + (size_t)n * D_DIM * HW;
  const int* seg = (n < B_IMG) ? (sp + (size_t)n * HW)
                               : (rsp + (size_t)(n - B_IMG) * HW);
  const int pix0 = blk * (TILES_PER_BLOCK * TILE_PX);

  const float* gmean = ws + OFF_MEAN + (size_t)n * C_SEG * D_DIM;
  const float* gcnt = ws + OFF_CNT + n * C_SEG;
  for (int i = t; i < C_SEG * D_DIM; i += TPB) {
    int c = i >> 5, d = i & 31;
    lmean[c * MSTRIDE + d] = gmean[i];
  }
  for (int i = t; i < C_SEG; i += TPB) lcnt[i] = gcnt[i];
  if (t == 0) lred = 0.0f;

  issue_tile(ebase, pix0, ltile[0], t);

  float acc = 0.0f;
  for (int tile = 0; tile < TILES_PER_BLOCK; ++tile) {
    if (tile + 1 < TILES_PER_BLOCK) {
      issue_tile(ebase, pix0 + (tile + 1) * TILE_PX, ltile[(tile + 1) & 1], t);
      WAIT_ASYNC(8);
    } else {
      WAIT_ASYNC(0);
    }
    __syncthreads();

    const float* buf = ltile[tile & 1];
    const int p = pix0 + tile * TILE_PX + t;
    const int s = seg[p];
    const float4* mrow = (const float4*)&lmean[s * MSTRIDE];
    float dot = 0.0f;
#pragma unroll
    for (int k = 0; k < 8; ++k) {
      float4 m4 = mrow[k];
      dot += m4.x * buf[(4 * k + 0) * TILE_PX + t]
           + m4.y * buf[(4 * k + 1) * TILE_PX + t]
           + m4.z * buf[(4 * k + 2) * TILE_PX + t]
           + m4.w * buf[(4 * k + 3) * TILE_PX + t];
    }
    acc += fmaxf(1.0f - 0.5f * dot - DELTA_VARF, 0.0f) / lcnt[s];
    __syncthreads();
  }

  atomicAdd(&lred, acc);
  __syncthreads();
  if (t == 0) atomicAdd(&ws[OFF_SCAL + n], lred);
}

// ---------------- kernel 4: edge loss via WMMA diagonal trick ---------------
// One wave per image. For 16 edges: A[e][k] = m[i_e][k], B[k][e] = m[j_e][k];
// diag(A x B) gives the 16 needed dots. K=16 via 4 chained 16x16x4 f32 WMMAs.
__global__ __launch_bounds__(32) void k_edges(const int* __restrict__ edges,
                                              const float* __restrict__ wts,
                                              float* __restrict__ ws) {
  const int b = blockIdx.x;        // image
  const int l = threadIdx.x;       // lane 0..31, full wave, EXEC all ones
  const int row = l & 15;
  const int koff = (l >> 4) ? 2 : 0;   // f32 A/B layout: hi half-lanes hold K+2
  float s = 0.0f;

  for (int h = 0; h < 2; ++h) {
    const float* m = ws + OFF_MEAN + (size_t)(h ? (B_IMG + b) : b) * C_SEG * D_DIM;
    for (int tile = 0; tile < E_EDGE / 16; ++tile) {
      const int base = tile * 16;
      const int ei = edges[(size_t)b * 2 * E_EDGE + base + row];
      const int ej = edges[(size_t)b * 2 * E_EDGE + E_EDGE + base + row];
      const float* mi = m + ei * D_DIM;
      const float* mj = m + ej * D_DIM;
      v8f acc = {0.f, 0.f, 0.f, 0.f, 0.f, 0.f, 0.f, 0.f};
#pragma unroll
      for (int kb = 0; kb < 4; ++kb) {
        const int k0 = 4 * kb + koff;
        v2f a, bb;
        a.x = mi[k0]; a.y = mi[k0 + 1];
        bb.x = mj[k0]; bb.y = mj[k0 + 1];
        acc = __builtin_amdgcn_wmma_f32_16x16x4_f32(
            false, a, false, bb, (short)0, acc, false, false);
      }
      // diagonal: element (e,e) lives in VGPR e%8, lane e (e<8) / lane e+16 (e>=8)
#pragma unroll
      for (int r = 0; r < 8; ++r) {
        const float dlo = __shfl(acc[r], r, 32);
        const float dhi = __shfl(acc[r], r + 24, 32);
        const float w0 = wts[(size_t)b * E_EDGE + base + r];
        const float w1 = wts[(size_t)b * E_EDGE + base + r + 8];
        s += fmaxf(DELTA_DISTF - (1.0f - dlo) * w0, 0.0f);
        s += fmaxf(DELTA_DISTF - (1.0f - dhi) * w1, 0.0f);
      }
    }
  }
  if (l == 0) ws[OFF_SCAL + 4 + b] = s * (1.0f / E_EDGE);
}

// ---------------- kernel 5: rd/rd2 cross terms + final combine --------------
__global__ __launch_bounds__(32) void k_final(const float* __restrict__ rot,
                                              float* __restrict__ ws,
                                              float* __restrict__ out) {
  const int l = threadIdx.x;   // lane = dim d
  float loss = 0.0f;
  for (int b = 0; b < B_IMG; ++b) {
    const float* ma = ws + OFF_MEAN + (size_t)b * C_SEG * D_DIM;
    const float* mb = ws + OFF_MEAN + (size_t)(B_IMG + b) * C_SEG * D_DIM;
    float t0 = 0.0f, t1 = 0.0f;
    for (int c = 0; c < 16; ++c) t0 += ma[c * D_DIM + l] * mb[c * D_DIM + l];
    for (int c = 16; c < 32; ++c) t1 += ma[c * D_DIM + l] * mb[c * D_DIM + l];
#pragma unroll
    for (int off = 16; off; off >>= 1) {
      t0 += __shfl_xor(t0, off, 32);
      t1 += __shfl_xor(t1, off, 32);
    }
    const float rd = 1.0f - t0 * (1.0f / 32.0f);
    const float rd2 = 1.0f - t1 * (1.0f / 32.0f);
    const float s1 = ws[OFF_SCAL + 4 + b] + fmaxf(rd - DELTA_VARF, 0.0f);
    const float s2 = fmaxf(DELTA_DISTF - rd2, 0.0f) * rot[0];
    const float intra = (ws[OFF_SCAL + b] + ws[OFF_SCAL + 2 + b]) * (1.0f / C_SEG);
    loss += (s1 + s2) + intra;   // ALPHA = BETA = 1
  }
  if (l == 0) out[0] = loss;
}

// ---------------- launcher ----------------
extern "C" void kernel_launch(void* const* d_in, const int* in_sizes, int n_in,
                              void* d_out, int out_size, void* d_ws, size_t ws_size,
                              hipStream_t stream) {
  const float* emb = (const float*)d_in[0];
  const int* sp = (const int*)d_in[1];
  const int* rsp = (const int*)d_in[2];
  const int* edges = (const int*)d_in[3];
  const float* wts = (const float*)d_in[4];
  const float* rot = (const float*)d_in[6];
  float* ws = (float*)d_ws;
  float* out = (float*)d_out;

  k_zero<<<(ZERO_FLOATS + TPB - 1) / TPB, TPB, 0, stream>>>(ws);
  k_segsum<<<NMAPS * BLOCKS_PER_MAP, TPB, 0, stream>>>(emb, sp, rsp, ws);
  k_means<<<(NMAPS * C_SEG + TPB - 1) / TPB, TPB, 0, stream>>>(ws);
  k_intra<<<NMAPS * BLOCKS_PER_MAP, TPB, 0, stream>>>(emb, sp, rsp, ws);
  k_edges<<<B_IMG, 32, 0, stream>>>(edges, wts, ws);
  k_final<<<1, 32, 0, stream>>>(rot, ws, out);
}